// ILQR_11785390260580
// MI455X (gfx1250) — compile-verified
//
#include <hip/hip_runtime.h>
#include <hip/hip_bf16.h>

// ---------------------------------------------------------------------------
// iLQR backward Riccati recursion for MI455X (gfx1250, wave32, WMMA).
// Single persistent workgroup; all per-step matrices live in LDS; every GEMM
// runs on V_WMMA_F32_16X16X4_F32 (fp32 matrix core — matches reference dtype).
// Split-K dual accumulators give two independent WMMA chains per tile
// (covers WMMA->WMMA RAW latency) with zero thread-dependent branching.
// ---------------------------------------------------------------------------

#define NX 128
#define NA 64
#define TSTEPS 1024
#define NS_ITERS 20          // Newton–Schulz iterations for Qaa^{-1}

typedef float v2f __attribute__((ext_vector_type(2)));
typedef float v8f __attribute__((ext_vector_type(8)));

// Element accessor with compile-time transpose.
template <bool TR>
__device__ __forceinline__ float mat_at(const float* p, int ld, int r, int c) {
    return TR ? p[c * ld + r] : p[r * ld + c];
}

// C (row-major MxN in LDS, ldc=N) = [Cinit] + op(A) * op(B).
// 16x16 tiles round-robin over wave32 waves. K reduction is split into two
// interleaved chains (chunks 0,8,16,... and 4,12,20,...) accumulated into
// independent registers, summed at the end. K must be a multiple of 8.
//
// Fragment layouts (CDNA5 ISA 7.12.2, fp32):
//   A 16x4 : lane<16 -> (A[m][k+0], A[m][k+1]); lane>=16 -> (A[m][k+2], A[m][k+3])
//   B 4x16 : lane<16 -> (B[k+0][n], B[k+1][n]); lane>=16 -> (B[k+2][n], B[k+3][n])
//   C 16x16: vgpr r, M = r + 8*(lane>>4), N = lane&15
template <bool TA, bool TB, bool HASC>
__device__ void gemm16(float* Csh, int M, int N, int K,
                       const float* Ap, int lda,
                       const float* Bp, int ldb,
                       const float* Cinit, int ldci) {
    const int lane = threadIdx.x & 31;
    const int wave = threadIdx.x >> 5;
    const int nw   = blockDim.x >> 5;
    const int half = lane >> 4;      // 0 or 1
    const int l15  = lane & 15;
    const int tM = M >> 4, tN = N >> 4;
    for (int tile = wave; tile < tM * tN; tile += nw) {
        const int tm = (tile / tN) << 4;
        const int tn = (tile % tN) << 4;
        v8f c0 = {}, c1 = {};
        if (HASC) {
#pragma unroll
            for (int r = 0; r < 8; ++r)
                c0[r] = Cinit[(tm + 8 * half + r) * ldci + tn + l15];
        }
        const int am = tm + l15;     // A row this lane supplies
        const int bn = tn + l15;     // B col this lane supplies
        for (int kk = 0; kk < K; kk += 8) {
            const int ka = kk + 2 * half;
            const int kb = kk + 4 + 2 * half;
            v2f a0, b0, a1, b1;
            a0.x = mat_at<TA>(Ap, lda, am, ka);
            a0.y = mat_at<TA>(Ap, lda, am, ka + 1);
            b0.x = mat_at<TB>(Bp, ldb, ka, bn);
            b0.y = mat_at<TB>(Bp, ldb, ka + 1, bn);
            a1.x = mat_at<TA>(Ap, lda, am, kb);
            a1.y = mat_at<TA>(Ap, lda, am, kb + 1);
            b1.x = mat_at<TB>(Bp, ldb, kb, bn);
            b1.y = mat_at<TB>(Bp, ldb, kb + 1, bn);
            c0 = __builtin_amdgcn_wmma_f32_16x16x4_f32(
                     false, a0, false, b0, (short)0, c0, false, false);
            c1 = __builtin_amdgcn_wmma_f32_16x16x4_f32(
                     false, a1, false, b1, (short)0, c1, false, false);
        }
#pragma unroll
        for (int r = 0; r < 8; ++r)
            Csh[(tm + 8 * half + r) * N + tn + l15] = c0[r] + c1[r];
    }
}

// Dynamic-LDS layout (float offsets). Regions reused by lifetime:
//   A [0,16384)      : Vxx (alive through Wx/Wa GEMMs)
//                      -> Qax(+0, 8192) | Qaa(+8192, 4096)
//                      overwritten last by the symmetrized new Vxx.
//   B [16384,32768)  : Wx / Wx+fx   ->  T_ns(+0,4096) | M2(+4096,8192)
//   C [32768,40960)  : Wa / Wa+fa   ->  Xa(+0,4096)   | Xb(+4096,4096)
//   D [40960,57344)  : Vacc = Qxx accumulator -> final (unsymmetrized) Vxx'
//   E [57344,65536)  : K (64x128)
//   F [65536,73728)  : staged fa_t (128x64)
//   V [73728,74368)  : small vectors
#define LDS_FLOATS 74368

__global__ __launch_bounds__(512) void ilqr_backward_kernel(
    const float* __restrict__ fx, const float* __restrict__ fa,
    const float* __restrict__ lx, const float* __restrict__ la,
    const float* __restrict__ lxx, const float* __restrict__ laa,
    const float* __restrict__ lax, float* __restrict__ out) {
    extern __shared__ float sm[];
    float* sVxx  = sm;                 // 128x128
    float* sQax  = sm;                 // 64x128  (aliases dead Vxx)
    float* sQaa  = sm + 8192;          // 64x64
    float* sWx   = sm + 16384;         // 128x128
    float* sTns  = sm + 16384;         // 64x64   (aliases dead Wx)
    float* sM2   = sm + 16384 + 4096;  // 64x128
    float* sWa   = sm + 32768;         // 128x64
    float* sXa   = sm + 32768;         // 64x64   (aliases dead Wa)
    float* sXb   = sm + 32768 + 4096;  // 64x64
    float* sVacc = sm + 40960;         // 128x128
    float* sK    = sm + 57344;         // 64x128
    float* sfa   = sm + 65536;         // 128x64 staged fa_t
    float* sVx   = sm + 73728;         // 128
    float* sQx   = sVx + 128;          // 128
    float* sQa   = sQx + 128;          // 64
    float* sk    = sQa + 64;           // 64
    float* st2   = sk + 64;            // 64
    float* sVxn  = st2 + 64;           // 128
    float* sred  = sVxn + 128;         // 64

    const int tid = threadIdx.x;
    const int nth = blockDim.x;

    // Terminal condition: Vx = lx[T-1], Vxx = lxx[T-1].
    for (int i = tid; i < NX * NX; i += nth)
        sVxx[i] = lxx[(size_t)(TSTEPS - 1) * NX * NX + i];
    for (int i = tid; i < NX; i += nth)
        sVx[i] = lx[(size_t)(TSTEPS - 1) * NX + i];
    __syncthreads();

    for (int t = TSTEPS - 1; t >= 0; --t) {
        const float* fx_t  = fx  + (size_t)t * NX * NX;
        const float* fa_t  = fa  + (size_t)t * NX * NA;
        const float* lx_t  = lx  + (size_t)t * NX;
        const float* la_t  = la  + (size_t)t * NA;
        const float* lxx_t = lxx + (size_t)t * NX * NX;
        const float* laa_t = laa + (size_t)t * NA * NA;
        const float* lax_t = lax + (size_t)t * NA * NX;

        // Stage fa_t into LDS (coalesced); prefetch next step's dynamics.
        for (int i = tid; i < NX * NA; i += nth) sfa[i] = fa_t[i];
        if (t > 0) {
            const float* nfx = fx_t - NX * NX;
            const float* nfa = fa_t - NX * NA;
            for (int i = tid; i < 256; i += nth) __builtin_prefetch(nfx + i * 64, 0, 1);
            for (int i = tid; i < 128; i += nth) __builtin_prefetch(nfa + i * 64, 0, 1);
        }
        __syncthreads();

        // P1: Qx = lx + fx^T Vx ; Qa = la + fa^T Vx
        if (tid < NX) {
            float s = lx_t[tid];
            for (int j = 0; j < NX; ++j) s += fx_t[j * NX + tid] * sVx[j];
            sQx[tid] = s;
        } else if (tid < NX + NA) {
            const int a = tid - NX;
            float s = la_t[a];
            for (int j = 0; j < NX; ++j) s += sfa[j * NA + a] * sVx[j];
            sQa[a] = s;
        }
        __syncthreads();

        // P2: Wx = Vxx @ fx ; Wa = Vxx @ fa
        gemm16<false, false, false>(sWx, NX, NX, NX, sVxx, NX, fx_t, NX, nullptr, 0);
        gemm16<false, false, false>(sWa, NX, NA, NX, sVxx, NX, sfa, NA, nullptr, 0);
        __syncthreads();

        // P3: Vacc = Qxx = lxx + fx^T @ Wx   (old Vxx now dead)
        gemm16<true, false, true>(sVacc, NX, NX, NX, fx_t, NX, sWx, NX, lxx_t, NX);
        __syncthreads();

        // P4: Wx += fx ; Wa += fa   (Vreg = Vxx + I, MU = 1)
        for (int i = tid; i < NX * NX; i += nth) sWx[i] += fx_t[i];
        for (int i = tid; i < NX * NA; i += nth) sWa[i] += sfa[i];
        __syncthreads();

        // P5: Qax = lax + fa^T @ (Wx+fx) ; Qaa = laa + fa^T @ (Wa+fa)
        gemm16<true, false, true>(sQax, NA, NX, NX, sfa, NA, sWx, NX, lax_t, NX);
        gemm16<true, false, true>(sQaa, NA, NA, NX, sfa, NA, sWa, NA, laa_t, NA);
        __syncthreads();

        // P6: Newton-Schulz init: X0 = Qaa / (||Qaa||_1 * ||Qaa||_inf)
        if (tid < NA) {
            float s = 0.f;
            for (int j = 0; j < NA; ++j) s += fabsf(sQaa[tid * NA + j]);
            sred[tid] = s;
        }
        __syncthreads();
        for (int s = 32; s > 0; s >>= 1) {
            if (tid < s) sred[tid] = fmaxf(sred[tid], sred[tid + s]);
            __syncthreads();
        }
        const float c0 = 1.0f / (sred[0] * sred[0]);   // symmetric: n1 == ninf
        for (int i = tid; i < NA * NA; i += nth) sXa[i] = c0 * sQaa[i];
        __syncthreads();

        // P7: X <- X (2I - Qaa X), quadratic convergence (Qaa SPD).
        float* Xc = sXa;
        float* Xo = sXb;
        for (int it = 0; it < NS_ITERS; ++it) {
            gemm16<false, false, false>(sTns, NA, NA, NA, sQaa, NA, Xc, NA, nullptr, 0);
            __syncthreads();
            gemm16<false, false, false>(Xo, NA, NA, NA, Xc, NA, sTns, NA, nullptr, 0);
            __syncthreads();
            for (int i = tid; i < NA * NA; i += nth) Xo[i] = 2.f * Xc[i] - Xo[i];
            __syncthreads();
            float* tmp = Xc; Xc = Xo; Xo = tmp;
        }
        // Qaa^{-1} ~= Xc ; Qaa_r == Qaa (clamp inactive for SPD Qaa).

        // P8: k = -Qaa_inv @ Qa  (waves 0-1), K = -(Qaa_inv @ Qax) (all waves)
        if (tid < NA) {
            float s = 0.f;
            for (int j = 0; j < NA; ++j) s += Xc[tid * NA + j] * sQa[j];
            sk[tid] = -s;
        }
        gemm16<false, false, false>(sK, NA, NX, NA, Xc, NA, sQax, NX, nullptr, 0);
        __syncthreads();
        for (int i = tid; i < NA * NX; i += nth) sK[i] = -sK[i];
        __syncthreads();

        // P9: t2 = Qa + Qaa@k ;  Vx' = Qx + K^T t2 + Qax^T k
        if (tid < NA) {
            float s = sQa[tid];
            for (int j = 0; j < NA; ++j) s += sQaa[tid * NA + j] * sk[j];
            st2[tid] = s;
        }
        __syncthreads();
        if (tid < NX) {
            float s = sQx[tid];
            for (int a = 0; a < NA; ++a)
                s += sK[a * NX + tid] * st2[a] + sQax[a * NX + tid] * sk[a];
            sVxn[tid] = s;
        }
        // P10: M2 = Qax + Qaa @ K
        gemm16<false, false, true>(sM2, NA, NX, NA, sQaa, NA, sK, NX, sQax, NX);
        __syncthreads();
        // P11: Vacc += K^T @ M2 ;  Vacc += Qax^T @ K
        gemm16<true, false, true>(sVacc, NX, NX, NA, sK, NX, sM2, NX, sVacc, NX);
        __syncthreads();
        gemm16<true, false, true>(sVacc, NX, NX, NA, sQax, NX, sK, NX, sVacc, NX);
        __syncthreads();

        // P12: emit gains[t] = [k | K]  (64 x 129)
        float* out_t = out + (size_t)t * NA * (NX + 1);
        for (int i = tid; i < NA * (NX + 1); i += nth) {
            const int a = i / (NX + 1), c = i % (NX + 1);
            out_t[i] = (c == 0) ? sk[a] : sK[a * NX + (c - 1)];
        }
        // P13: Vxx = 0.5 (Vacc + Vacc^T) ; Vx = Vx'   (overwrites Qax/Qaa region)
        for (int i = tid; i < NX * NX; i += nth) {
            const int r = i >> 7, c = i & (NX - 1);
            sVxx[i] = 0.5f * (sVacc[i] + sVacc[c * NX + r]);
        }
        if (tid < NX) sVx[tid] = sVxn[tid];
        __syncthreads();
    }
}

extern "C" void kernel_launch(void* const* d_in, const int* in_sizes, int n_in,
                              void* d_out, int out_size, void* d_ws, size_t ws_size,
                              hipStream_t stream) {
    const float* fx  = (const float*)d_in[0];
    const float* fa  = (const float*)d_in[1];
    const float* lx  = (const float*)d_in[2];
    const float* la  = (const float*)d_in[3];
    const float* lxx = (const float*)d_in[4];
    const float* laa = (const float*)d_in[5];
    const float* lax = (const float*)d_in[6];
    float* out = (float*)d_out;

    const size_t lds_bytes = (size_t)LDS_FLOATS * sizeof(float);
    hipFuncSetAttribute((const void*)ilqr_backward_kernel,
                        hipFuncAttributeMaxDynamicSharedMemorySize, (int)lds_bytes);
    ilqr_backward_kernel<<<1, 512, lds_bytes, stream>>>(fx, fa, lx, la, lxx, laa,
                                                        lax, out);
}